// BilinearInteraction_8993661517937
// MI455X (gfx1250) — compile-verified
//
#include <hip/hip_runtime.h>

#define BATCH 8192
#define NF    40
#define ED    32
#define NK    780   // 40*39/2

typedef float v2f __attribute__((ext_vector_type(2)));
typedef float v8f __attribute__((ext_vector_type(8)));

// Butterfly merge: combine two per-lane value streams (x for lanes with
// (lane&mask)==0, y for the others) while reducing across `mask`.
// Returns, on mask==0 lanes: x(me)+x(partner); on mask==1 lanes: y(me)+y(partner).
__device__ __forceinline__ float bmerge(float x, float y, int mask, int lane) {
  float keep = (lane & mask) ? y : x;
  float send = (lane & mask) ? x : y;
  return keep + __shfl_xor(send, mask, 32);
}

// out[b,k] = sum_{e,f} emb[b,i_k,e] * W[k,e,f] * emb[b,j_k,f]
//
// Per wave: one interaction k, 16-batch tiles.
//   T(16x32) = Vj(16x32) @ W[k]^T  via 16x WMMA_F32_16X16X4_F32 (two e-halves, 8 K-steps)
//   out[b]   = sum_e Vi[b,e] * T[b,e]  via an 8-shuffle butterfly-merge reduction.
__global__ __launch_bounds__(256) void bilinear_wmma_kernel(
    const float* __restrict__ emb,   // (8192, 40, 32)
    const float* __restrict__ W,     // (780, 32, 32)
    float* __restrict__ out) {       // (8192, 780)
  const int k = blockIdx.x;

  // Recover (i, j) = triu_indices(40, k=1)[k]  (uniform -> scalar loop)
  int i = 0, rem = k;
  while (rem >= (NF - 1 - i)) { rem -= (NF - 1 - i); ++i; }
  const int j = i + 1 + rem;

  const int lane = threadIdx.x & 31;
  const int wave = threadIdx.x >> 5;
  const int l16  = lane & 15;
  const int half = lane >> 4;          // which 16-lane half

  const float* Wk = W + (size_t)k * (ED * ED);

  // Pre-format W[k] as WMMA B operands (B is 4x16, K x N with N = e).
  // B[f_rel, e] = W[k, e, f]; lane(l16,half) holds pair f = 4s + 2*half, +1
  // for row e = 16h + l16. Held in 32 VGPRs across all batch tiles.
  v2f bop[2][8];
#pragma unroll
  for (int h = 0; h < 2; ++h)
#pragma unroll
    for (int s = 0; s < 8; ++s)
      bop[h][s] = *(const v2f*)(Wk + (16 * h + l16) * ED + 4 * s + 2 * half);

  const int gw = blockIdx.y * 8 + wave;   // 0..127 waves per k
#pragma unroll 1
  for (int t = 0; t < 4; ++t) {
    const int b0 = (gw * 4 + t) * 16;     // 16-batch tile base

    // A operand: Vj tile, A[m, f] = emb[b0+m, j, f].
    // lane(l16,half), K-step s: contiguous pair at f = 4s + 2*half.
    const float* vjrow = emb + ((size_t)(b0 + l16) * NF + j) * ED;

    v8f c0 = {};  // T tile, e = 0..15
    v8f c1 = {};  // T tile, e = 16..31
#pragma unroll
    for (int s = 0; s < 8; ++s) {
      v2f a = *(const v2f*)(vjrow + 4 * s + 2 * half);
      c0 = __builtin_amdgcn_wmma_f32_16x16x4_f32(false, a, false, bop[0][s],
                                                 (short)0, c0, false, false);
      c1 = __builtin_amdgcn_wmma_f32_16x16x4_f32(false, a, false, bop[1][s],
                                                 (short)0, c1, false, false);
    }

    // D layout: VGPR r, lane(l16,half) = T[m = r + 8*half, e = l16 (+16 for c1)].
    // p[r] = Vi[m, l16]*T[m, l16] + Vi[m, 16+l16]*T[m, 16+l16]
    float p[8];
#pragma unroll
    for (int r = 0; r < 8; ++r) {
      const float* vir = emb + ((size_t)(b0 + r + 8 * half) * NF + i) * ED;
      p[r] = c0[r] * vir[l16] + c1[r] * vir[16 + l16];
    }

    // Butterfly-merge tree: 8 values x 16 lanes -> 8 sums in 8 shuffles.
    float q0 = bmerge(p[0], p[1], 1, lane);
    float q1 = bmerge(p[2], p[3], 1, lane);
    float q2 = bmerge(p[4], p[5], 1, lane);
    float q3 = bmerge(p[6], p[7], 1, lane);
    float r0 = bmerge(q0, q1, 2, lane);
    float r1 = bmerge(q2, q3, 2, lane);
    float s0 = bmerge(r0, r1, 4, lane);
    float fin = s0 + __shfl_xor(s0, 8, 32);
    // Lane l16 now holds out for m = (l16 & 7) + 8*half (duplicated at l16^8).

    if (l16 < 8)
      out[(size_t)(b0 + l16 + 8 * half) * NK + k] = fin;
  }
}

extern "C" void kernel_launch(void* const* d_in, const int* in_sizes, int n_in,
                              void* d_out, int out_size, void* d_ws, size_t ws_size,
                              hipStream_t stream) {
  const float* emb = (const float*)d_in[0];   // (8192, 40, 32) fp32
  const float* W   = (const float*)d_in[1];   // (780, 32, 32) fp32
  float* out = (float*)d_out;                 // (8192, 780) fp32

  dim3 grid(NK, 16);          // 780 interactions x 16 batch-tile groups
  dim3 block(256);            // 8 wave32s; each wave: 4 tiles of 16 batches
  bilinear_wmma_kernel<<<grid, block, 0, stream>>>(emb, W, out);
}